// HGNNEncoder_10574209483281
// MI455X (gfx1250) — compile-verified
//
#include <hip/hip_runtime.h>
#include <math.h>

// ---- problem constants (from reference) ----
#define NNODES   3712
#define NROI     116
#define NGRAPH   32
#define NFEAT    128
#define NHID     64
#define NHEADS   4
#define ALPHA_LR 0.1f
#define A_PARA   0.1f
#define PHI_C    0.94247779607693793f      // 0.3*pi
#define MAXNORM  (1.0f - 4e-3f)

typedef __attribute__((ext_vector_type(2))) float v2f;
typedef __attribute__((ext_vector_type(8))) float v8f;

// D = A(16x4,f32) * B(4x16,f32) + C(16x16,f32) -- exact fp32 WMMA
__device__ __forceinline__ v8f wmma4f32(v2f a, v2f b, v8f c) {
  return __builtin_amdgcn_wmma_f32_16x16x4_f32(
      false, a, false, b, (short)0, c, false, false);
}

// fkernel = logmap0(project(x,1),1): pure per-row scalar rescale.
// r = ||x||.  Returns s with fkernel(x) = s*x, and note ||fkernel(x)|| = s*r.
__device__ __forceinline__ float fkernel_scale(float r) {
  float proj = 1.0f, rn = r;
  if (r > MAXNORM) { proj = MAXNORM / r; rn = MAXNORM; }
  rn = fmaxf(rn, 1e-15f);
  float t = fminf(rn, 1.0f - 1e-7f);
  return proj * (atanhf(t) / rn);
}

__device__ __forceinline__ float act_elu_cos(float v) {
  float e = (v > 0.0f) ? v : expm1f(v);
  return e + A_PARA * cosf(v + PHI_C);
}

// ---------------------------------------------------------------------------
// C[h] = A @ B[h]   (A: MxK row-major shared across heads; B[h]: KxNc; C[h]: MxNc)
// One wave per 16x16 output tile; K-loop step 4 with V_WMMA_F32_16X16X4_F32.
// M % 16 == 0, K % 4 == 0, Nc % 16 == 0 (3712/128|256/64 all satisfy).
// ---------------------------------------------------------------------------
__global__ __launch_bounds__(32)
void gemm_wmma_kernel(const float* __restrict__ A, const float* __restrict__ B,
                      float* __restrict__ C, int M, int K, int Nc) {
  const int h  = blockIdx.y;
  const float* Bh = B + (size_t)h * K * Nc;
  float*       Ch = C + (size_t)h * M * Nc;
  const int ntiles = Nc >> 4;
  const int mt = blockIdx.x / ntiles;
  const int nt = blockIdx.x % ntiles;
  const int l  = threadIdx.x;
  const int lm = l & 15;        // M (A) / N (B,D) lane index
  const int lh = l >> 4;        // half-wave select

  v8f acc = {0.f,0.f,0.f,0.f,0.f,0.f,0.f,0.f};
  const float* arow = A  + (size_t)(mt * 16 + lm) * K + 2 * lh;      // A[m][k0+2lh+v]
  const float* bcol = Bh + (size_t)(2 * lh) * Nc + nt * 16 + lm;     // B[k0+2lh+v][n]

  for (int k0 = 0; k0 < K; k0 += 4) {
    v2f a = { arow[k0],              arow[k0 + 1] };
    v2f b = { bcol[(size_t)k0 * Nc], bcol[(size_t)(k0 + 1) * Nc] };
    acc = wmma4f32(a, b, acc);
  }
  float* crow = Ch + (size_t)(mt * 16 + 8 * lh) * Nc + nt * 16 + lm; // D[v+8lh][n]
#pragma unroll
  for (int v = 0; v < 8; ++v) crow[(size_t)v * Nc] = acc[v];
}

// ---------------------------------------------------------------------------
// el[h][i] = Wh[h][i,:] . a[h][0:64],  er[h][i] = Wh[h][i,:] . a[h][64:128]
// ---------------------------------------------------------------------------
__global__ void attvec_kernel(const float* __restrict__ Wh,
                              const float* __restrict__ avec,
                              float* __restrict__ el, float* __restrict__ er,
                              int heads) {
  int idx = blockIdx.x * blockDim.x + threadIdx.x;
  if (idx >= heads * NNODES) return;
  int h = idx / NNODES, i = idx % NNODES;
  const float* row = Wh + ((size_t)h * NNODES + i) * NHID;
  const float* a1  = avec + (size_t)h * 2 * NHID;
  const float* a2  = a1 + NHID;
  float s1 = 0.f, s2 = 0.f;
#pragma unroll 4
  for (int k = 0; k < NHID; ++k) { s1 += row[k] * a1[k]; s2 += row[k] * a2[k]; }
  el[idx] = s1;
  er[idx] = s2;
}

// ---------------------------------------------------------------------------
// Per (graph block, head): masked softmax attention over the 116x116 diagonal
// block, hp = att @ Wh_block via f32 WMMA from LDS, fkernel (x2 for layer 2),
// elu + 0.1*cos activation, store to out[row][h*64 + c].
// blockDim = 256 (8 waves). Dynamic LDS: att 128x132 + whs 128x68 = 100 KB.
// ---------------------------------------------------------------------------
#define ATT_S 132
#define WHS_S 68
#define LDS_BYTES ((128 * ATT_S + 128 * WHS_S) * sizeof(float))

template <bool DOUBLE_FK>
__global__ __launch_bounds__(256)
void att_block_kernel(const float* __restrict__ adj, const float* __restrict__ Wh,
                      const float* __restrict__ el, const float* __restrict__ er,
                      float* __restrict__ out, int out_stride) {
  extern __shared__ float lds[];
  float* att = lds;                 // [128][ATT_S], zero-padded rows/cols >=116
  float* whs = lds + 128 * ATT_S;   // [128][WHS_S], rows >=116 zero

  const int g = blockIdx.x, h = blockIdx.y;
  const int t = threadIdx.x;
  const int r0 = g * NROI;

  for (int i = t; i < 128 * ATT_S; i += 256) att[i] = 0.f;
  for (int i = t; i < 128 * WHS_S; i += 256) whs[i] = 0.f;
  __syncthreads();

  // Stage Wh block (116x64) into LDS.
  const float* WhH = Wh + (size_t)h * NNODES * NHID;
  for (int i = t; i < NROI * NHID; i += 256) {
    int r = i >> 6, c = i & 63;
    whs[r * WHS_S + c] = WhH[(size_t)(r0 + r) * NHID + c];
  }

  // Masked softmax, one thread per row.
  const float* elh = el + (size_t)h * NNODES + r0;
  const float* erh = er + (size_t)h * NNODES + r0;
  if (t < NROI) {
    const float* adjrow = adj + (size_t)(r0 + t) * NNODES + r0;
    float* arow = att + t * ATT_S;
    const float eli = elh[t];
    float m = -INFINITY;
    for (int j = 0; j < NROI; ++j) {
      float e = -INFINITY;
      if (adjrow[j] > 0.f) {
        float z = eli + erh[j];
        e = (z > 0.f) ? z : ALPHA_LR * z;
        m = fmaxf(m, e);
      }
      arow[j] = e;
    }
    if (m == -INFINITY) {
      for (int j = 0; j < NROI; ++j) arow[j] = 0.f;
    } else {
      float s = 0.f;
      for (int j = 0; j < NROI; ++j) {
        float w = (arow[j] == -INFINITY) ? 0.f : expf(arow[j] - m);
        arow[j] = w; s += w;
      }
      float inv = (s > 0.f) ? 1.f / s : 0.f;
      for (int j = 0; j < NROI; ++j) arow[j] *= inv;
    }
  }
  __syncthreads();

  // hp = att(128x128, padded) @ whs(128x64) -- f32 WMMA, EXEC all ones here.
  const int w  = t >> 5;            // wave id -> M tile
  const int l  = t & 31;
  const int lm = l & 15, lh = l >> 4;
  v8f acc[4];
#pragma unroll
  for (int nt = 0; nt < 4; ++nt)
    acc[nt] = (v8f){0.f,0.f,0.f,0.f,0.f,0.f,0.f,0.f};

  const float* arowA = att + (w * 16 + lm) * ATT_S + 2 * lh;
  const float* bbase = whs + (2 * lh) * WHS_S + lm;
  for (int k0 = 0; k0 < 128; k0 += 4) {
    v2f a = { arowA[k0], arowA[k0 + 1] };
#pragma unroll
    for (int nt = 0; nt < 4; ++nt) {
      const float* bc = bbase + k0 * WHS_S + nt * 16;
      v2f b = { bc[0], bc[WHS_S] };
      acc[nt] = wmma4f32(a, b, acc[nt]);
    }
  }
  __syncthreads();                  // all waves done reading whs/att

  // Spill hp to LDS (reuse att region, stride WHS_S) for row-wise fkernel.
  float* hp = att;
#pragma unroll
  for (int nt = 0; nt < 4; ++nt)
#pragma unroll
    for (int v = 0; v < 8; ++v)
      hp[(w * 16 + v + 8 * lh) * WHS_S + nt * 16 + lm] = acc[nt][v];
  __syncthreads();

  if (t < NROI) {
    const float* row = hp + t * WHS_S;
    float ss = 0.f;
    for (int c = 0; c < NHID; ++c) ss += row[c] * row[c];
    float r = sqrtf(ss);
    float s = fkernel_scale(r);
    if (DOUBLE_FK) s *= fkernel_scale(s * r);   // ||fkernel(x)|| = s*r
    float* orow = out + (size_t)(r0 + t) * out_stride + h * NHID;
    for (int c = 0; c < NHID; ++c) orow[c] = act_elu_cos(s * row[c]);
  }
}

// ---------------------------------------------------------------------------
extern "C" void kernel_launch(void* const* d_in, const int* in_sizes, int n_in,
                              void* d_out, int out_size, void* d_ws, size_t ws_size,
                              hipStream_t stream) {
  (void)in_sizes; (void)n_in; (void)out_size; (void)ws_size;
  const float* x       = (const float*)d_in[0];   // [3712,128]
  const float* adj     = (const float*)d_in[1];   // [3712,3712]
  const float* W_heads = (const float*)d_in[2];   // [4,128,64]
  const float* a_heads = (const float*)d_in[3];   // [4,128,1]
  const float* W_out   = (const float*)d_in[4];   // [256,64]
  const float* a_out   = (const float*)d_in[5];   // [128,1]
  float* out = (float*)d_out;                     // [3712,64]

  float* ws = (float*)d_ws;
  float* Wh   = ws;                  ws += (size_t)NHEADS * NNODES * NHID; // 950272
  float* el1  = ws;                  ws += (size_t)NHEADS * NNODES;
  float* er1  = ws;                  ws += (size_t)NHEADS * NNODES;
  float* xcat = ws;                  ws += (size_t)NNODES * NHEADS * NHID; // 950272
  float* Wh2  = ws;                  ws += (size_t)NNODES * NHID;
  float* el2  = ws;                  ws += (size_t)NNODES;
  float* er2  = ws;                  /* end */

  const int mtiles = NNODES / 16;    // 232

  // Layer 1: Wh[h] = x @ W_heads[h]
  gemm_wmma_kernel<<<dim3(mtiles * (NHID / 16), NHEADS), 32, 0, stream>>>(
      x, W_heads, Wh, NNODES, NFEAT, NHID);
  attvec_kernel<<<(NHEADS * NNODES + 255) / 256, 256, 0, stream>>>(
      Wh, a_heads, el1, er1, NHEADS);
  att_block_kernel<false><<<dim3(NGRAPH, NHEADS), 256, LDS_BYTES, stream>>>(
      adj, Wh, el1, er1, xcat, NHEADS * NHID);

  // Layer 2: Wh2 = xcat @ W_out
  gemm_wmma_kernel<<<dim3(mtiles * (NHID / 16), 1), 32, 0, stream>>>(
      xcat, W_out, Wh2, NNODES, NHEADS * NHID, NHID);
  attvec_kernel<<<(NNODES + 255) / 256, 256, 0, stream>>>(
      Wh2, a_out, el2, er2, 1);
  att_block_kernel<true><<<dim3(NGRAPH, 1), 256, LDS_BYTES, stream>>>(
      adj, Wh2, el2, er2, out, NHID);
}